// MolecularGCNEncoder_11897059410622
// MI455X (gfx1250) — compile-verified
//
#include <hip/hip_runtime.h>
#include <hip/hip_bf16.h>

typedef __attribute__((ext_vector_type(2))) float v2f;
typedef __attribute__((ext_vector_type(8))) float v8f;

#define HID 64
#define EPS 1e-5f

// ---------------- utility ----------------
__global__ void fill_f32(float* __restrict__ p, float v, int n) {
    int t = blockIdx.x * blockDim.x + threadIdx.x;
    if (t < n) p[t] = v;
}

// ---------------- degree / norm ----------------
__global__ void deg_kernel(const long long* __restrict__ dst, float* __restrict__ deg, int nedges) {
    int t = blockIdx.x * blockDim.x + threadIdx.x;
    if (t < nedges) atomicAdd(&deg[(int)dst[t]], 1.0f);
}

__global__ void rsqrt_inplace(float* __restrict__ d, int n) {
    int t = blockIdx.x * blockDim.x + threadIdx.x;
    if (t < n) d[t] = rsqrtf(d[t]);   // deg >= 1 always (self loop)
}

// ---------------- WMMA f32 GEMM: Y[M,64] = X[M,K] @ W[K,64] ----------------
// One wave (32 lanes) computes one 16x16 tile of Y using V_WMMA_F32_16X16X4_F32.
__global__ void gemm_wmma_f32(const float* __restrict__ X, const float* __restrict__ W,
                              float* __restrict__ Y, int M, int K) {
    int gwave = (blockIdx.x * blockDim.x + threadIdx.x) >> 5;
    int lane  = threadIdx.x & 31;
    int mtile = gwave >> 2;          // 64/16 = 4 N-tiles
    int ntile = gwave & 3;
    if (mtile * 16 >= M) return;

    int row   = mtile * 16 + (lane & 15);  // A: lane%16 = M row (both halves)
    int col   = ntile * 16 + (lane & 15);  // B/C/D: lane%16 = N col
    int khalf = (lane >> 4) * 2;           // lanes 0-15: K={0,1}; lanes 16-31: K={2,3}

    v8f acc = {};
    for (int k0 = 0; k0 < K; k0 += 4) {
        int kk = k0 + khalf;
        v2f a, b;
        a.x = X[row * K + kk];
        a.y = X[row * K + kk + 1];
        b.x = W[kk * HID + col];
        b.y = W[(kk + 1) * HID + col];
        acc = __builtin_amdgcn_wmma_f32_16x16x4_f32(
            /*neg_a=*/false, a, /*neg_b=*/false, b,
            /*c_mod=*/(short)0, acc, /*reuse_a=*/false, /*reuse_b=*/false);
    }
    // C/D layout: VGPR r -> M = r (lanes 0-15) or r+8 (lanes 16-31)
    int orow = mtile * 16 + ((lane >> 4) << 3);
#pragma unroll
    for (int r = 0; r < 8; ++r)
        Y[(orow + r) * HID + col] = acc[r];
}

// ---------------- self-loop + bias (initializes agg, no pre-zero needed) ----------------
__global__ void selfloop_bias(const float* __restrict__ tmp, const float* __restrict__ dis,
                              const float* __restrict__ b, float* __restrict__ agg, int n) {
    int t = blockIdx.x * blockDim.x + threadIdx.x;
    if (t >= n * HID) return;
    int node = t >> 6, f = t & 63;
    float d = dis[node];
    agg[t] = tmp[t] * d * d + b[f];
}

// ---------------- edge scatter: agg[dst] += tmp[src] * dis[src]*dis[dst] ----------------
// 16 threads per edge, float4 gather, 4 float atomics each.
__global__ void scatter_edges(const long long* __restrict__ src, const long long* __restrict__ dst,
                              const float* __restrict__ dis, const float* __restrict__ tmp,
                              float* __restrict__ agg, int nedges) {
    int t = blockIdx.x * blockDim.x + threadIdx.x;
    int e = t >> 4;
    if (e >= nedges) return;
    int fg = (t & 15) << 2;
    int s = (int)src[e], d = (int)dst[e];
    float nrm = dis[s] * dis[d];
    const float4 v = *reinterpret_cast<const float4*>(tmp + s * HID + fg);
    float* base = agg + d * HID + fg;
    atomicAdd(base + 0, v.x * nrm);
    atomicAdd(base + 1, v.y * nrm);
    atomicAdd(base + 2, v.z * nrm);
    atomicAdd(base + 3, v.w * nrm);
}

// ---------------- BatchNorm: per-feature sum / sumsq ----------------
#define BN_CHUNK 512
__global__ void bn_stats(const float* __restrict__ agg, float* __restrict__ stats, int n) {
    int f   = threadIdx.x & 63;
    int sub = threadIdx.x >> 6;                 // 4 parallel node sub-streams
    int end = min(blockIdx.x * BN_CHUNK + BN_CHUNK, n);
    float s = 0.f, s2 = 0.f;
    for (int nd = blockIdx.x * BN_CHUNK + sub; nd < end; nd += 4) {
        float v = agg[nd * HID + f];
        s += v; s2 += v * v;
    }
    atomicAdd(&stats[f], s);
    atomicAdd(&stats[HID + f], s2);
}

__global__ void bn_finalize(const float* __restrict__ g, const float* __restrict__ be,
                            float* __restrict__ stats, float invn) {
    int f = threadIdx.x;
    if (f >= HID) return;
    float mu  = stats[f] * invn;
    float var = stats[HID + f] * invn - mu * mu;
    float sc  = g[f] * rsqrtf(var + EPS);
    stats[128 + f] = sc;
    stats[192 + f] = be[f] - mu * sc;
}

__global__ void bn_apply_relu(float* __restrict__ h, const float* __restrict__ stats, int n) {
    int t = blockIdx.x * blockDim.x + threadIdx.x;
    if (t >= n * HID) return;
    int f = t & 63;
    h[t] = fmaxf(0.f, h[t] * stats[128 + f] + stats[192 + f]);
}

// ---------------- global mean pool ----------------
__global__ void pool_cnt(const long long* __restrict__ batch, float* __restrict__ cnt, int n) {
    int t = blockIdx.x * blockDim.x + threadIdx.x;
    if (t < n) atomicAdd(&cnt[(int)batch[t]], 1.0f);
}

__global__ void pool_sum(const long long* __restrict__ batch, const float* __restrict__ h,
                         float* __restrict__ out, int n) {
    int t = blockIdx.x * blockDim.x + threadIdx.x;
    if (t >= n * HID) return;
    int node = t >> 6, f = t & 63;
    atomicAdd(&out[(int)batch[node] * HID + f], h[t]);
}

__global__ void pool_div(float* __restrict__ out, const float* __restrict__ cnt, int ngraphs) {
    int t = blockIdx.x * blockDim.x + threadIdx.x;
    if (t >= ngraphs * HID) return;
    out[t] /= fmaxf(cnt[t >> 6], 1.0f);
}

// ---------------- host orchestration ----------------
extern "C" void kernel_launch(void* const* d_in, const int* in_sizes, int n_in,
                              void* d_out, int out_size, void* d_ws, size_t ws_size,
                              hipStream_t stream) {
    const float*     x     = (const float*)d_in[0];
    const long long* eidx  = (const long long*)d_in[1];
    const long long* batch = (const long long*)d_in[2];
    const float* W[3] = { (const float*)d_in[3], (const float*)d_in[7],  (const float*)d_in[11] };
    const float* b[3] = { (const float*)d_in[4], (const float*)d_in[8],  (const float*)d_in[12] };
    const float* g[3] = { (const float*)d_in[5], (const float*)d_in[9],  (const float*)d_in[13] };
    const float* be[3]= { (const float*)d_in[6], (const float*)d_in[10], (const float*)d_in[14] };

    const int N   = in_sizes[0] / 128;     // 80000 nodes, NODE_DIM=128
    const int E   = in_sizes[1] / 2;       // 1280000 edges
    const int NG  = out_size / HID;        // 2048 graphs
    const long long* esrc = eidx;
    const long long* edst = eidx + E;

    // workspace layout (floats): dis | A (h/agg) | B (tmp) | stats(256) | cnt
    float* ws   = (float*)d_ws;
    float* dis  = ws;                      // N
    float* A    = dis + N;                 // N*64  (agg, then h in-place)
    float* B    = A + (size_t)N * HID;     // N*64  (gemm output)
    float* stats= B + (size_t)N * HID;     // 256
    float* cnt  = stats + 256;             // NG
    float* out  = (float*)d_out;

    const int BLK = 256;
    auto blocks = [](long long n, int blk) { return (int)((n + blk - 1) / blk); };

    // --- normalization: deg -> dis ---
    fill_f32<<<blocks(N, BLK), BLK, 0, stream>>>(dis, 1.0f, N);      // self loop
    deg_kernel<<<blocks(E, BLK), BLK, 0, stream>>>(edst, dis, E);
    rsqrt_inplace<<<blocks(N, BLK), BLK, 0, stream>>>(dis, N);

    const int mtiles     = (N + 15) / 16;
    const int gemm_blks  = blocks((long long)mtiles * 4 * 32, BLK);
    const int elem_blks  = blocks((long long)N * HID, BLK);
    const int edge_blks  = blocks((long long)E * 16, BLK);
    const int stat_blks  = (N + BN_CHUNK - 1) / BN_CHUNK;
    const float invn     = 1.0f / (float)N;

    // --- 3 GCN layers ---
    for (int L = 0; L < 3; ++L) {
        const float* in = (L == 0) ? x : A;
        const int    K  = (L == 0) ? 128 : HID;
        gemm_wmma_f32<<<gemm_blks, BLK, 0, stream>>>(in, W[L], B, N, K);
        selfloop_bias<<<elem_blks, BLK, 0, stream>>>(B, dis, b[L], A, N);
        scatter_edges<<<edge_blks, BLK, 0, stream>>>(esrc, edst, dis, B, A, E);
        fill_f32<<<1, 128, 0, stream>>>(stats, 0.0f, 128);
        bn_stats<<<stat_blks, BLK, 0, stream>>>(A, stats, N);
        bn_finalize<<<1, 64, 0, stream>>>(g[L], be[L], stats, invn);
        bn_apply_relu<<<elem_blks, BLK, 0, stream>>>(A, stats, N);
    }

    // --- global mean pool ---
    fill_f32<<<blocks((long long)NG * HID, BLK), BLK, 0, stream>>>(out, 0.0f, NG * HID);
    fill_f32<<<blocks(NG, BLK), BLK, 0, stream>>>(cnt, 0.0f, NG);
    pool_cnt<<<blocks(N, BLK), BLK, 0, stream>>>(batch, cnt, N);
    pool_sum<<<elem_blks, BLK, 0, stream>>>(batch, A, out, N);
    pool_div<<<blocks((long long)NG * HID, BLK), BLK, 0, stream>>>(out, cnt, NG);
}